// L1Butterfly_3143916060796
// MI455X (gfx1250) — compile-verified
//
#include <hip/hip_runtime.h>
#include <cstdint>

#define WIDTH   1024
#define HALF    512
#define DEPTH   10
#define BATCH   32768
#define TPB     512          // one thread per column pair, x2 rows in flight
#define NBLOCKS 2048         // 2 rows/iter -> 8 iterations per block

// ---------------------------------------------------------------------------
// Async stage TWO consecutive 1024-float rows (8 KB) into an LDS buffer using
// the CDNA5 async global->LDS engine (ASYNCcnt). All 512 threads move 16 B.
// Generic pointers to LDS carry the LDS byte offset in their low 32 bits.
// ---------------------------------------------------------------------------
__device__ __forceinline__ void async_stage_rows(const float* __restrict__ X,
                                                 int rowBase, float* ldsBuf, int t) {
    const float* g = X + (size_t)rowBase * WIDTH + 4 * (size_t)t;
    uint32_t lds   = (uint32_t)(uintptr_t)(ldsBuf + 4 * t);
    asm volatile("global_load_async_to_lds_b128 %0, %1, off"
                 :: "v"(lds), "v"(g) : "memory");
}

__device__ __forceinline__ void wait_async_done() {
    asm volatile("s_wait_asynccnt 0" ::: "memory");
}

// One L1 rotation. Float path: 2 cmp + 1 abs-add + 2 cndmask (fold, using the
// identity fold(x,y) = (|x|,|y|) or (|y|,|x|)) + 3 fma + 1 min (reflection
// as y3 = min(y2, 2r-y2)) + 4 cndmask (unfold). Quadrant bits stay in i1
// masks on the scalar pipe.
__device__ __forceinline__ void l1rot(float x, float y, float m,
                                      bool ib0, bool ib1,
                                      float& xo, float& yo) {
    bool  xp = (x >= 0.0f);
    bool  yp = (y >= 0.0f);
    float ax = fabsf(x);
    float ay = fabsf(y);
    float r  = ax + ay;

    bool qb0 = (xp != yp);        // q = 2*!yp + (xp^yp)
    bool qb1 = !yp;

    // fold into quadrant 0: {xq,yq} = {|x|,|y|}, swapped when qb0
    float xq = qb0 ? ay : ax;
    float yq = qb0 ? ax : ay;

    // shear along the L1 circle, reflect at the corner via min
    float x2 = fmaf(-m, r, xq);
    float y2 = fmaf( m, r, yq);
    float y3 = fminf(y2, fmaf(2.0f, r, -y2));

    // c2 = (q + floor(c)) & 3 as a 2-bit add in i1 masks (SALU co-issue)
    bool cb0   = (qb0 != ib0);
    bool carry = (qb0 && ib0);
    bool cb1   = ((qb1 != ib1) != carry);

    // unfold: rotate +90 if cb0, then 180 if cb1
    float xb = cb0 ? -y3 : x2;
    float yb = cb0 ?  x2 : y3;
    xo = cb1 ? -xb : xb;
    yo = cb1 ? -yb : yb;
}

__global__ __launch_bounds__(TPB) void
l1_butterfly_kernel(const float* __restrict__ X,
                    const float* __restrict__ P,
                    float* __restrict__ Y) {
    __shared__ float smem[3 * 2 * WIDTH];   // triple buffer of 2-row chunks: 24 KB
    const int t = threadIdx.x;              // pair index, shared by both rows

    // ---- Per-block (row-invariant) param precompute -----------------------
    float c0v[DEPTH];
    bool  ib0[DEPTH], ib1[DEPTH];
#pragma unroll
    for (int i = 0; i < DEPTH; ++i) {
        float cc  = P[t * DEPTH + i];
        float fl  = floorf(cc);
        c0v[i]    = cc - fl;                // fractional part (c in [0,4))
        int   icc = (int)fl;
        ib0[i]    = (icc & 1) != 0;
        ib1[i]    = (icc & 2) != 0;
    }

    // Rotating triple-buffer pointers (each buffer holds 2 rows).
    float* bufA = smem;                     // staged current 2-row chunk
    float* bufB = smem + 2 * WIDTH;         // async prefetch target
    float* bufC = smem + 4 * WIDTH;         // ping-pong partner

    const int base0 = 2 * blockIdx.x;
    async_stage_rows(X, base0, bufA, t);

    for (int base = base0; base < BATCH; base += 2 * gridDim.x) {
        // Chunk ready: drain ASYNCcnt, then make it visible to all 16 waves.
        wait_async_done();
        __syncthreads();

        // Prefetch the next 2-row chunk into an untouched buffer.
        int nextBase = base + 2 * gridDim.x;
        if (nextBase < BATCH) async_stage_rows(X, nextBase, bufB, t);

#pragma unroll
        for (int layer = 0; layer < DEPTH; ++layer) {
            float* src = (layer & 1) ? bufC : bufA;   // even layers read bufA
            // Two independent chains (row0/row1) -> VOPD dual-issue fodder.
            float x0 = src[t];                        // ds_load_2addr_stride64
            float y0 = src[t + HALF];
            float x1 = src[t + WIDTH];
            float y1 = src[t + WIDTH + HALF];

            float xo0, yo0, xo1, yo1;
            float m = c0v[layer];
            l1rot(x0, y0, m, ib0[layer], ib1[layer], xo0, yo0);
            l1rot(x1, y1, m, ib0[layer], ib1[layer], xo1, yo1);

            if (layer < DEPTH - 1) {
                // Perfect shuffle: out[2p]=nx0[p], out[2p+1]=nx1[p].
                float* dst = (layer & 1) ? bufA : bufC;   // dual b64 stores
                *(float2*)(dst + 2 * t)         = make_float2(xo0, yo0);
                *(float2*)(dst + WIDTH + 2 * t) = make_float2(xo1, yo1);
                __syncthreads();
            } else {
                // Final layer: interleave straight to global, coalesced b64.
                *(float2*)(Y + (size_t)base * WIDTH + 2 * t)       = make_float2(xo0, yo0);
                *(float2*)(Y + (size_t)(base + 1) * WIDTH + 2 * t) = make_float2(xo1, yo1);
            }
        }

        // Rotate buffers: prefetched chunk becomes current; old current
        // becomes ping-pong partner; old partner becomes prefetch target.
        float* tmp = bufA;
        bufA = bufB;
        bufB = bufC;
        bufC = tmp;
    }
}

extern "C" void kernel_launch(void* const* d_in, const int* in_sizes, int n_in,
                              void* d_out, int out_size, void* d_ws, size_t ws_size,
                              hipStream_t stream) {
    const float* X = (const float*)d_in[0];   // (32768, 1024) f32
    const float* P = (const float*)d_in[1];   // (512, 10) f32
    float*       Y = (float*)d_out;           // (32768, 1024) f32
    (void)in_sizes; (void)n_in; (void)out_size; (void)d_ws; (void)ws_size;

    hipLaunchKernelGGL(l1_butterfly_kernel, dim3(NBLOCKS), dim3(TPB), 0, stream,
                       X, P, Y);
}